// RelativeSelfAttention_46239617908978
// MI455X (gfx1250) — compile-verified
//
#include <hip/hip_runtime.h>

typedef float v2f __attribute__((ext_vector_type(2)));
typedef float v8f __attribute__((ext_vector_type(8)));

#define NHEADS 16
#define HD     64
#define SEQ    2048
#define DM     1024
#define NBH    32          // b * h
#define RWIN   128
#define RSZ    257         // 2*RWIN + 1
#define RPST   260         // Srp row stride (floats)
#define TST    66          // 64-wide tile row stride (floats), even -> 8B-aligned float2 frags

__device__ __forceinline__ v8f wmma4(v2f a, v2f b, v8f c) {
    // D = A(16x4,f32) * B(4x16,f32) + C(16x16,f32)
    return __builtin_amdgcn_wmma_f32_16x16x4_f32(false, a, false, b, (short)0, c, false, false);
}

// =====================================================================
// Kernel 1: qkv[4096,3072] = X[4096,1024] @ Wqkv^T + bqkv
// scattered into q/k/v workspaces laid out [b][h][s][hd]
// =====================================================================
__global__ __launch_bounds__(256, 1)
void qkv_gemm(const float* __restrict__ X, const float* __restrict__ W,
              const float* __restrict__ bias,
              float* __restrict__ qws, float* __restrict__ kws, float* __restrict__ vws)
{
    __shared__ float As[128 * 18];   // A tile [m][k], stride 18
    __shared__ float Bs[128 * 18];   // B tile stored [n][k], stride 18

    const int tid  = threadIdx.x;
    const int lane = tid & 31;
    const int wave = tid >> 5;
    const int lo   = lane & 15;
    const int hi   = lane >> 4;
    const int wm   = (wave & 3) * 32;   // wave row offset
    const int wn   = (wave >> 2) * 64;  // wave col offset
    const int mB   = blockIdx.x * 128;
    const int nB   = blockIdx.y * 128;

    const v8f zv = {0.f, 0.f, 0.f, 0.f, 0.f, 0.f, 0.f, 0.f};
    v8f acc[2][4];
    #pragma unroll
    for (int mt = 0; mt < 2; ++mt)
        #pragma unroll
        for (int nt = 0; nt < 4; ++nt)
            acc[mt][nt] = zv;

    const int lrow = tid >> 1;          // 0..127
    const int lk8  = (tid & 1) << 3;    // 0 or 8

    for (int ks = 0; ks < DM; ks += 16) {
        __syncthreads();
        {
            const float* ga = X + (size_t)(mB + lrow) * DM + ks + lk8;
            float4 a0 = *(const float4*)ga;
            float4 a1 = *(const float4*)(ga + 4);
            float* da = As + lrow * 18 + lk8;
            da[0] = a0.x; da[1] = a0.y; da[2] = a0.z; da[3] = a0.w;
            da[4] = a1.x; da[5] = a1.y; da[6] = a1.z; da[7] = a1.w;

            const float* gb = W + (size_t)(nB + lrow) * DM + ks + lk8;
            float4 b0 = *(const float4*)gb;
            float4 b1 = *(const float4*)(gb + 4);
            float* db = Bs + lrow * 18 + lk8;
            db[0] = b0.x; db[1] = b0.y; db[2] = b0.z; db[3] = b0.w;
            db[4] = b1.x; db[5] = b1.y; db[6] = b1.z; db[7] = b1.w;
        }
        __syncthreads();

        #pragma unroll
        for (int s4 = 0; s4 < 4; ++s4) {
            const int k0 = (s4 << 2) + (hi << 1);
            v2f a0 = *(const v2f*)(As + (wm + lo) * 18 + k0);
            v2f a1 = *(const v2f*)(As + (wm + 16 + lo) * 18 + k0);
            #pragma unroll
            for (int nt = 0; nt < 4; ++nt) {
                v2f b = *(const v2f*)(Bs + (wn + nt * 16 + lo) * 18 + k0);
                acc[0][nt] = wmma4(a0, b, acc[0][nt]);
                acc[1][nt] = wmma4(a1, b, acc[1][nt]);
            }
        }
    }

    // epilogue: add bias, scatter into q/k/v [b][h][s][hd]
    #pragma unroll
    for (int nt = 0; nt < 4; ++nt) {
        const int col    = nB + wn + nt * 16 + lo;       // 0..3071
        const float bv   = bias[col];
        const int head   = col / 192;
        const int within = col - head * 192;
        const int which  = within >> 6;                  // 0=q 1=k 2=v
        const int dim    = within & 63;
        float* dst = (which == 0) ? qws : (which == 1) ? kws : vws;
        #pragma unroll
        for (int mt = 0; mt < 2; ++mt) {
            #pragma unroll
            for (int i = 0; i < 8; ++i) {
                const int row   = mB + wm + mt * 16 + i + (hi << 3);  // 0..4095
                const int batch = row >> 11;
                const int si    = row & 2047;
                dst[(((size_t)batch * NHEADS + head) * SEQ + si) * HD + dim] =
                    acc[mt][nt][i] + bv;
            }
        }
    }
}

// =====================================================================
// Kernel 2: flash attention with relative-position bias.
// One WG (4 waves) per (b*h, 64-query tile). rp tile lives in LDS.
// =====================================================================
__global__ __launch_bounds__(128, 1)
void rel_attn(const float* __restrict__ qws, const float* __restrict__ kws,
              const float* __restrict__ vws, const float* __restrict__ Wr,
              const float* __restrict__ brv, float* __restrict__ out)
{
    extern __shared__ float sm[];
    float* Sq  = sm;                    // [64][TST]
    float* Sk  = Sq + 64 * TST;         // [64][TST]
    float* Sv  = Sk + 64 * TST;         // [64][TST]
    float* Sp  = Sv + 64 * TST;         // [64][TST]
    float* Srp = Sp + 64 * TST;         // [64][RPST], 257 valid cols

    const int bh   = blockIdx.x;        // 0..31
    const int qt   = blockIdx.y;        // 0..31
    const int tid  = threadIdx.x;
    const int lane = tid & 31;
    const int wave = tid >> 5;          // 0..3, owns query rows [16w,16w+16)
    const int lo   = lane & 15;
    const int hi   = lane >> 4;

    const int batch = bh >> 4;
    const int head  = bh & 15;
    const size_t base = (size_t)bh * SEQ * HD;
    const int q0 = qt * 64;

    // ---- stage Q tile ----
    for (int idx = tid; idx < 64 * 16; idx += 128) {
        const int r  = idx >> 4;
        const int c4 = (idx & 15) << 2;
        float4 v = *(const float4*)(qws + base + (size_t)(q0 + r) * HD + c4);
        float* d = Sq + r * TST + c4;
        d[0] = v.x; d[1] = v.y; d[2] = v.z; d[3] = v.w;
    }
    __syncthreads();

    // ---- rp = Q_tile @ Wr^T + br  -> Srp[64][257] ----
    for (int idx = tid; idx < 64 * RSZ; idx += 128) {
        const int r = idx / RSZ;
        const int o = idx - r * RSZ;
        const float* wr = Wr + o * HD;
        const float* qq = Sq + r * TST;
        float s = 0.f;
        #pragma unroll 8
        for (int k = 0; k < HD; ++k) s += qq[k] * wr[k];
        Srp[r * RPST + o] = s + brv[o];
    }

    const v8f zv = {0.f, 0.f, 0.f, 0.f, 0.f, 0.f, 0.f, 0.f};
    v8f oacc[4];
    #pragma unroll
    for (int nt = 0; nt < 4; ++nt) oacc[nt] = zv;
    float mrun[8], lrun[8];
    #pragma unroll
    for (int i = 0; i < 8; ++i) { mrun[i] = -__builtin_inff(); lrun[i] = 0.f; }

    __syncthreads();   // Srp ready; Sk/Sv free for staging

    for (int kt = 0; kt < 32; ++kt) {
        // ---- stage K/V tiles ----
        for (int idx = tid; idx < 64 * 16; idx += 128) {
            const int r  = idx >> 4;
            const int c4 = (idx & 15) << 2;
            const size_t g = base + (size_t)(kt * 64 + r) * HD + c4;
            float4 kv = *(const float4*)(kws + g);
            float4 vv = *(const float4*)(vws + g);
            float* dk = Sk + r * TST + c4;
            float* dv = Sv + r * TST + c4;
            dk[0] = kv.x; dk[1] = kv.y; dk[2] = kv.z; dk[3] = kv.w;
            dv[0] = vv.x; dv[1] = vv.y; dv[2] = vv.z; dv[3] = vv.w;
        }
        __syncthreads();

        // ---- scores S = Q K^T (per wave: 16 rows x 64 cols) ----
        v8f sc[4];
        #pragma unroll
        for (int nt = 0; nt < 4; ++nt) sc[nt] = zv;
        #pragma unroll
        for (int ksub = 0; ksub < 16; ++ksub) {
            const int k0 = (ksub << 2) + (hi << 1);
            v2f a = *(const v2f*)(Sq + (wave * 16 + lo) * TST + k0);
            #pragma unroll
            for (int nt = 0; nt < 4; ++nt) {
                v2f b = *(const v2f*)(Sk + (nt * 16 + lo) * TST + k0);
                sc[nt] = wmma4(a, b, sc[nt]);
            }
        }

        // ---- scale + gathered relative-position bias; row max ----
        float rmax[8];
        #pragma unroll
        for (int i = 0; i < 8; ++i) rmax[i] = -__builtin_inff();
        #pragma unroll
        for (int nt = 0; nt < 4; ++nt) {
            const int jg = kt * 64 + nt * 16 + lo;
            #pragma unroll
            for (int i = 0; i < 8; ++i) {
                const int rt = wave * 16 + i + (hi << 3);
                const int ig = q0 + rt;
                int off = jg - ig;
                off = (off < -RWIN) ? -RWIN : (off > RWIN) ? RWIN : off;
                float s = sc[nt][i] * 0.125f + Srp[rt * RPST + off + RWIN];
                sc[nt][i] = s;
                rmax[i] = fmaxf(rmax[i], s);
            }
        }
        #pragma unroll
        for (int d = 1; d < 16; d <<= 1)
            #pragma unroll
            for (int i = 0; i < 8; ++i)
                rmax[i] = fmaxf(rmax[i], __shfl_xor(rmax[i], d));

        // ---- online softmax ----
        float mnew[8], alpha[8], rsum[8];
        #pragma unroll
        for (int i = 0; i < 8; ++i) {
            mnew[i]  = fmaxf(mrun[i], rmax[i]);
            alpha[i] = __expf(mrun[i] - mnew[i]);
            mrun[i]  = mnew[i];
            rsum[i]  = 0.f;
        }
        #pragma unroll
        for (int nt = 0; nt < 4; ++nt)
            #pragma unroll
            for (int i = 0; i < 8; ++i) {
                float p = __expf(sc[nt][i] - mnew[i]);
                sc[nt][i] = p;
                rsum[i] += p;
            }
        #pragma unroll
        for (int d = 1; d < 16; d <<= 1)
            #pragma unroll
            for (int i = 0; i < 8; ++i)
                rsum[i] += __shfl_xor(rsum[i], d);
        #pragma unroll
        for (int i = 0; i < 8; ++i) lrun[i] = lrun[i] * alpha[i] + rsum[i];

        // rescale running output
        #pragma unroll
        for (int nt = 0; nt < 4; ++nt)
            #pragma unroll
            for (int i = 0; i < 8; ++i)
                oacc[nt][i] *= alpha[i];

        // ---- P -> LDS (C-layout scatter), then O += P V via WMMA ----
        #pragma unroll
        for (int nt = 0; nt < 4; ++nt)
            #pragma unroll
            for (int i = 0; i < 8; ++i)
                Sp[(wave * 16 + i + (hi << 3)) * TST + nt * 16 + lo] = sc[nt][i];
        __syncthreads();

        #pragma unroll
        for (int ksub = 0; ksub < 16; ++ksub) {
            const int k0 = (ksub << 2) + (hi << 1);
            v2f a = *(const v2f*)(Sp + (wave * 16 + lo) * TST + k0);
            #pragma unroll
            for (int nt = 0; nt < 4; ++nt) {
                v2f b;
                b.x = Sv[k0 * TST + nt * 16 + lo];
                b.y = Sv[(k0 + 1) * TST + nt * 16 + lo];
                oacc[nt] = wmma4(a, b, oacc[nt]);
            }
        }
        __syncthreads();   // done with Sk/Sv/Sp before next staging
    }

    // ---- normalize and write out[b][s][h*64+dim] ----
    #pragma unroll
    for (int nt = 0; nt < 4; ++nt) {
        const int dim = nt * 16 + lo;
        #pragma unroll
        for (int i = 0; i < 8; ++i) {
            const int row = q0 + wave * 16 + i + (hi << 3);
            out[((size_t)batch * SEQ + row) * DM + head * HD + dim] =
                oacc[nt][i] / lrun[i];
        }
    }
}

extern "C" void kernel_launch(void* const* d_in, const int* in_sizes, int n_in,
                              void* d_out, int out_size, void* d_ws, size_t ws_size,
                              hipStream_t stream)
{
    const float* x    = (const float*)d_in[0];  // [2,2048,1024]
    const float* Wqkv = (const float*)d_in[1];  // [3072,1024]
    const float* bqkv = (const float*)d_in[2];  // [3072]
    const float* Wr   = (const float*)d_in[3];  // [257,64]
    const float* br   = (const float*)d_in[4];  // [257]
    float* out = (float*)d_out;

    float* ws  = (float*)d_ws;
    const size_t per = (size_t)NBH * SEQ * HD;  // 4 M floats each
    float* qws = ws;
    float* kws = ws + per;
    float* vws = ws + 2 * per;

    dim3 g1(32, 24), b1(256);
    qkv_gemm<<<g1, b1, 0, stream>>>(x, Wqkv, bqkv, qws, kws, vws);

    dim3 g2(NBH, SEQ / 64), b2(128);
    size_t smem = (size_t)(4 * 64 * TST + 64 * RPST) * sizeof(float);  // ~134 KB
    rel_attn<<<g2, b2, smem, stream>>>(qws, kws, vws, Wr, br, out);
}